// Local2d_47390669144707
// MI455X (gfx1250) — compile-verified
//
#include <hip/hip_runtime.h>

// ---------------------------------------------------------------------------
// Local2d (untied-weight conv), B=32, C_in=C_out=64, H=W=32, k=3, pad=1.
// Per output position (x,y): D[64x32] = W[64x576] * P[576x32]  (fp32).
// Weight stream (151 MB, read once) dominates -> HBM-bound (~7 us @ 23.3 TB/s).
// V_WMMA_F32_16X16X4_F32; one block per position; 4 waves; each wave owns one
// 16-row M tile x full N=32 (2 accumulators share 1 A fragment) so every
// weight element is fetched from HBM exactly once.
//
// LDS holds the patch matrix TRANSPOSED, P_t[n][k], row stride 580 floats:
//  - B fragment = 2 contiguous floats -> single ds_load_b64, no re-pairing movs
//  - 580 mod 64 == 4: lane t hits banks {4t,4t+1}, lane t+16 {4t+2,4t+3}
//    -> one b64 request touches all 64 banks exactly once (conflict-free)
// ---------------------------------------------------------------------------

typedef __attribute__((ext_vector_type(2))) float v2f;
typedef __attribute__((ext_vector_type(8))) float v8f;

#define C_IN   64
#define C_OUT  64
#define HW     32
#define K_TOT  576           // C_IN * 3 * 3
#define LDS_T  580           // padded row stride of P_t[n][k]

__global__ __launch_bounds__(128, 1)
void local2d_wmma_f32(const float* __restrict__ xin,
                      const float* __restrict__ wgt,
                      const float* __restrict__ bias,
                      float* __restrict__ out)
{
    const int pos = blockIdx.x;        // 0..1023 output position
    const int x0  = pos >> 5;
    const int y0  = pos & 31;

    __shared__ float lds[32 * LDS_T];  // 72.5 KB: P_t[n][k]

    const int tid = threadIdx.x;       // 0..127

    // ---- stage P_t[n][k] = x[b=n, i, x0+dk-1, y0+dl-1] into LDS ----
    // consecutive threads -> consecutive k -> contiguous, conflict-free writes
    for (int e = tid; e < 32 * K_TOT; e += 128) {
        const int n  = e / K_TOT;
        const int k  = e - n * K_TOT;
        const int i  = k / 9;
        const int r9 = k - i * 9;
        const int dk = r9 / 3;
        const int dl = r9 - dk * 3;
        const int h  = x0 + dk - 1;
        const int w  = y0 + dl - 1;
        float v = 0.0f;
        if ((unsigned)h < HW && (unsigned)w < HW)
            v = xin[(((n * C_IN) + i) * HW + h) * HW + w];
        lds[n * LDS_T + k] = v;
    }
    __syncthreads();

    const int lane  = tid & 31;
    const int wave  = tid >> 5;                 // 0..3 -> M tile
    const int m0    = wave * 16;
    const int mrow  = m0 + (lane & 15);         // A-fragment row for this lane
    const int khalf = (lane >> 4) * 2;          // lanes 16-31 hold K+2,K+3

    const float* wbase = wgt + (size_t)pos * (C_OUT * K_TOT);
    const float* arow  = wbase + mrow * K_TOT + khalf;

    // Warm the weight stream (straight-line, outside the hot loop; keeps the
    // gfx1250 global_prefetch path without perturbing loop structure).
    __builtin_prefetch(arow + 128, 0, 3);
    __builtin_prefetch(arow + 320, 0, 3);

    // LDS base offsets (in floats) for the two B fragments of this lane
    const int bbase0 = (lane & 15) * LDS_T + khalf;        // N = 0..15
    const int bbase1 = bbase0 + 16 * LDS_T;                // N = 16..31

    v8f c0 = {};   // N = 0..15
    v8f c1 = {};   // N = 16..31

    #pragma unroll 4
    for (int k0 = 0; k0 < K_TOT; k0 += 4) {
        // A fragment (16x4 f32): lane L holds W[mrow][k0 + khalf + {0,1}]
        v2f a = *(const v2f*)(arow + k0);

        // B fragments (4x16 f32): contiguous pair -> ds_load_b64
        v2f b0 = *(const v2f*)(lds + bbase0 + k0);
        v2f b1 = *(const v2f*)(lds + bbase1 + k0);

        c0 = __builtin_amdgcn_wmma_f32_16x16x4_f32(
                 false, a, false, b0, (short)0, c0, false, false);
        c1 = __builtin_amdgcn_wmma_f32_16x16x4_f32(
                 false, a, false, b1, (short)0, c1, false, false);
    }

    // ---- store: C/D VGPR r holds M = m0 + r + 8*(lane/16), N = lane%16 ----
    const int mo   = m0 + (lane >> 4) * 8;
    const int n_lo = lane & 15;
    #pragma unroll
    for (int r = 0; r < 8; ++r) {
        const int m  = mo + r;
        const float bv = bias[m * (HW * HW) + pos];
        out[((n_lo        * C_OUT) + m) * (HW * HW) + pos] = c0[r] + bv;
        out[(((n_lo + 16) * C_OUT) + m) * (HW * HW) + pos] = c1[r] + bv;
    }
}

extern "C" void kernel_launch(void* const* d_in, const int* in_sizes, int n_in,
                              void* d_out, int out_size, void* d_ws, size_t ws_size,
                              hipStream_t stream) {
    const float* xin  = (const float*)d_in[0];   // (32, 64, 32, 32)
    const float* wgt  = (const float*)d_in[1];   // (32, 32, 64, 64, 3, 3)
    const float* bias = (const float*)d_in[2];   // (64, 32, 32)
    float* out = (float*)d_out;                  // (32, 64, 32, 32)

    dim3 grid(HW * HW);   // 1024 blocks: one per output position
    dim3 block(128);      // 4 waves (wave32)
    hipLaunchKernelGGL(local2d_wmma_f32, grid, block, 0, stream,
                       xin, wgt, bias, out);
}